// GCNGraphEmbedding_7773890806280
// MI455X (gfx1250) — compile-verified
//
#include <hip/hip_runtime.h>
#include <hip/hip_bf16.h>

typedef __attribute__((ext_vector_type(2))) float v2f;
typedef __attribute__((ext_vector_type(8))) float v8f;

// ---------------------------------------------------------------------------
// dinv[i] = 1.0f  (self-loop contribution to degree)
__global__ void set_one_kernel(float* __restrict__ p, int n) {
    int i = blockIdx.x * blockDim.x + threadIdx.x;
    if (i < n) p[i] = 1.0f;
}

// deg accumulation over edge destinations
__global__ void deg_kernel(const int* __restrict__ dst, float* __restrict__ deg, int nE) {
    int e = blockIdx.x * blockDim.x + threadIdx.x;
    if (e < nE) atomicAdd(&deg[dst[e]], 1.0f);
}

// deg -> rsqrt(deg) in place (deg >= 1 always, since self-loops are counted)
__global__ void rsqrt_kernel(float* __restrict__ p, int n) {
    int i = blockIdx.x * blockDim.x + threadIdx.x;
    if (i < n) {
        float d = p[i];
        p[i] = (d > 0.0f) ? rsqrtf(d) : 0.0f;
    }
}

// ---------------------------------------------------------------------------
// C[n,128] = A[n,128] @ W[128,128] via V_WMMA_F32_16X16X4_F32.
// Block = 256 threads = 8 waves; each block owns a 16-row strip, each wave a
// 16-wide column tile. A strip staged through LDS (stride 132 to avoid bank
// conflicts); W rows read through L2 (64 KB, fully L2 resident).
// Epilogue: bounce C tile through LDS, then one coalesced b128 write-out.
__global__ void gemm128_wmma(const float* __restrict__ A,
                             const float* __restrict__ W,
                             float* __restrict__ C,
                             int n) {
    __shared__ float As[16 * 132];
    const int block_row = blockIdx.x * 16;
    const int tid  = threadIdx.x;
    const int wave = tid >> 5;
    const int lane = tid & 31;

    // cooperative load of the 16x128 A strip (coalesced)
    for (int i = tid; i < 16 * 128; i += 256) {
        int r = i >> 7, c = i & 127;
        int gr = block_row + r;
        As[r * 132 + c] = (gr < n) ? A[(size_t)gr * 128 + c] : 0.0f;
    }
    __syncthreads();

    const int n0   = wave * 16;      // this wave's output column tile
    const int half = lane >> 4;      // 0: K pair {k0,k0+1}, 1: {k0+2,k0+3}
    const int l    = lane & 15;

    v8f acc = {};
    #pragma unroll 4
    for (int k0 = 0; k0 < 128; k0 += 4) {
        const int ka = k0 + half * 2;
        v2f a, b;
        // A fragment: 16x4 tile, lane l holds row M=l, K=ka..ka+1
        a.x = As[l * 132 + ka];
        a.y = As[l * 132 + ka + 1];
        // B fragment: 4x16 tile, lane l holds col N=n0+l, K=ka..ka+1
        b.x = W[(size_t)ka       * 128 + n0 + l];
        b.y = W[(size_t)(ka + 1) * 128 + n0 + l];
        acc = __builtin_amdgcn_wmma_f32_16x16x4_f32(
            /*neg_a=*/false, a, /*neg_b=*/false, b,
            /*c_mod=*/(short)0, acc, /*reuse_a=*/false, /*reuse_b=*/false);
    }

    // transpose-through-LDS epilogue: C/D layout is VGPR r, lanes 0-15 -> M=r,
    // lanes 16-31 -> M=r+8.  Reuse As (all reads of it are complete).
    __syncthreads();
    const int col = n0 + l;
    #pragma unroll
    for (int r = 0; r < 8; ++r) {
        As[(half * 8 + r) * 132 + col] = acc[r];
    }
    __syncthreads();

    // coalesced write-out: 16 rows x 128 cols as float4 (global_store_b128)
    for (int i = tid; i < 16 * 32; i += 256) {
        int r  = i >> 5;        // row within strip
        int c4 = (i & 31) * 4;  // float4 column offset
        int row = block_row + r;
        if (row < n) {
            const float* s = &As[r * 132 + c4];
            float4 v = make_float4(s[0], s[1], s[2], s[3]);
            *(float4*)(C + (size_t)row * 128 + c4) = v;
        }
    }
}

// ---------------------------------------------------------------------------
// One wave32 per edge: acc[dst] += hw[src] * dinv[src]*dinv[dst]
// Each lane moves a float4 (32 lanes * 4 = 128 channels).
__global__ void edge_agg_kernel(const float* __restrict__ hw,
                                const float* __restrict__ dinv,
                                const int* __restrict__ src,
                                const int* __restrict__ dst,
                                float* __restrict__ acc,
                                int nE) {
    int gwave = (blockIdx.x * blockDim.x + threadIdx.x) >> 5;
    int lane  = threadIdx.x & 31;
    if (gwave >= nE) return;
    int s = src[gwave];
    int d = dst[gwave];
    float norm = dinv[s] * dinv[d];
    const float4 v = ((const float4*)(hw + (size_t)s * 128))[lane];
    float* ap = acc + (size_t)d * 128 + lane * 4;
    atomicAdd(ap + 0, v.x * norm);
    atomicAdd(ap + 1, v.y * norm);
    atomicAdd(ap + 2, v.z * norm);
    atomicAdd(ap + 3, v.w * norm);
}

// acc = relu(acc + hw * dinv^2)   (adds the self-loop message, then ReLU)
__global__ void finalize_relu_kernel(float* __restrict__ acc,
                                     const float* __restrict__ hw,
                                     const float* __restrict__ dinv,
                                     int n) {
    int t = blockIdx.x * blockDim.x + threadIdx.x;
    if (t >= n * 128) return;
    int i = t >> 7;
    float di = dinv[i];
    float v = acc[t] + hw[t] * di * di;
    acc[t] = v > 0.0f ? v : 0.0f;
}

// Segment-max pool of h = [x1 | x2] over batch. Post-ReLU values are >= 0, so
// unsigned-int atomicMax on the raw bits is order-preserving; out is
// zero-initialized (== 0.0f, the identity for non-empty non-negative segments).
__global__ void pool_max_kernel(const float* __restrict__ x1,
                                const float* __restrict__ x2,
                                const int* __restrict__ batch,
                                unsigned int* __restrict__ out,
                                int n) {
    int t = blockIdx.x * blockDim.x + threadIdx.x;
    if (t >= n * 128) return;
    int i = t >> 7, c = t & 127;
    int g = batch[i];
    atomicMax(out + (size_t)g * 256 + c,       __float_as_uint(x1[t]));
    atomicMax(out + (size_t)g * 256 + 128 + c, __float_as_uint(x2[t]));
}

// ---------------------------------------------------------------------------
extern "C" void kernel_launch(void* const* d_in, const int* in_sizes, int n_in,
                              void* d_out, int out_size, void* d_ws, size_t ws_size,
                              hipStream_t stream) {
    const float* x     = (const float*)d_in[0];   // [N,128]
    const int*   eidx  = (const int*)  d_in[1];   // [2,E]
    const int*   batch = (const int*)  d_in[2];   // [N]
    const float* W1    = (const float*)d_in[3];   // [128,128]
    const float* W2    = (const float*)d_in[4];   // [128,128]

    const int nE = in_sizes[1] / 2;               // 800000
    const int nN = in_sizes[2];                   // 50000
    const int* src = eidx;
    const int* dst = eidx + nE;

    // workspace layout
    char* ws = (char*)d_ws;
    const size_t featBytes = (size_t)nN * 128 * sizeof(float);
    size_t off = ((size_t)nN * sizeof(float) + 255) & ~(size_t)255;
    float* dinv = (float*)ws;
    float* hw   = (float*)(ws + off);                  // shared GEMM output buffer
    float* a1   = (float*)(ws + off + featBytes);      // layer-1 accum -> x1
    float* a2   = (float*)(ws + off + 2 * featBytes);  // layer-2 accum -> x2

    hipMemsetAsync(a1, 0, featBytes, stream);
    hipMemsetAsync(a2, 0, featBytes, stream);
    hipMemsetAsync(d_out, 0, (size_t)out_size * sizeof(float), stream);

    const int B = 256;
    // degree -> dinv
    set_one_kernel<<<(nN + B - 1) / B, B, 0, stream>>>(dinv, nN);
    deg_kernel    <<<(nE + B - 1) / B, B, 0, stream>>>(dst, dinv, nE);
    rsqrt_kernel  <<<(nN + B - 1) / B, B, 0, stream>>>(dinv, nN);

    const int gemmBlocks = (nN + 15) / 16;
    const int edgeBlocks = (nE * 32 + B - 1) / B;       // one wave per edge
    const int featThreads = nN * 128;
    const int featBlocks  = (featThreads + B - 1) / B;

    // ---- layer 1: x1 = relu(Â (x @ W1)) ----
    gemm128_wmma        <<<gemmBlocks, B, 0, stream>>>(x, W1, hw, nN);
    edge_agg_kernel     <<<edgeBlocks, B, 0, stream>>>(hw, dinv, src, dst, a1, nE);
    finalize_relu_kernel<<<featBlocks, B, 0, stream>>>(a1, hw, dinv, nN);

    // ---- layer 2: x2 = relu(Â (x1 @ W2)) ----
    gemm128_wmma        <<<gemmBlocks, B, 0, stream>>>(a1, W2, hw, nN);
    edge_agg_kernel     <<<edgeBlocks, B, 0, stream>>>(hw, dinv, src, dst, a2, nE);
    finalize_relu_kernel<<<featBlocks, B, 0, stream>>>(a2, hw, dinv, nN);

    // ---- JK-cat + global max pool ----
    pool_max_kernel<<<featBlocks, B, 0, stream>>>(a1, a2, batch, (unsigned int*)d_out, nN);
}